// ReliabilityDiagram_10531259810372
// MI455X (gfx1250) — compile-verified
//
#include <hip/hip_runtime.h>
#include <stdint.h>

#define N_BINS 10
#define NCLASS 128
#define TILE_ROWS 48                       // 48 rows * 512B = 24KB per buffer (x2 buffers)
#define BLOCK_THREADS 256
#define WAVES_PER_BLOCK (BLOCK_THREADS / 32)
#define ROWS_PER_WAVE (TILE_ROWS / WAVES_PER_BLOCK)   // 6 rows per wave per tile

typedef float        f4    __attribute__((ext_vector_type(4)));
typedef unsigned int u32x4 __attribute__((ext_vector_type(4)));
typedef int          i32x8 __attribute__((ext_vector_type(8)));

// ---------------------------------------------------------------------------
// TDM issue: one descriptor moves a [TILE_ROWS x 128] f32 tile global -> LDS.
// D# packing per CDNA5 ISA (groups 0/1; 2D => groups 2/3
// omitted via the 2-operand instruction form).
// ---------------------------------------------------------------------------
__device__ __forceinline__ void tdm_issue(const float* sm, long long tile,
                                          uint32_t lds_base, int bufidx) {
    const uint64_t gaddr = (uint64_t)(uintptr_t)(sm + (size_t)tile * (TILE_ROWS * NCLASS));
    const uint32_t laddr = lds_base + (uint32_t)bufidx * (TILE_ROWS * NCLASS * 4);

    u32x4 g0;
    g0.x = 1u;                                                // count=1, user mode
    g0.y = laddr;                                             // lds_addr [63:32]
    g0.z = (uint32_t)gaddr;                                   // global_addr[31:0]
    g0.w = ((uint32_t)(gaddr >> 32) & 0x01FFFFFFu)            // global_addr[56:32]
         | (2u << 30);                                        // type=2 ("image")

    i32x8 g1;
    g1[0] = (2 << 16);                                        // data_size=2 (4B); no mask/pad/iter
    g1[1] = (NCLASS & 0xFFFF) << 16;                          // tensor_dim0[15:0] = 128
    g1[2] = (TILE_ROWS & 0xFFFF) << 16;                       // tensor_dim1[15:0] = 48
    g1[3] = (NCLASS & 0xFFFF) << 16;                          // tile_dim0 = 128
    g1[4] = TILE_ROWS;                                        // tile_dim1 = 48, tile_dim2 = 0
    g1[5] = NCLASS;                                           // tensor_dim0_stride[31:0] = 128
    g1[6] = ((TILE_ROWS * NCLASS) & 0xFFFF) << 16;            // dim1_stride[15:0]
    g1[7] = (TILE_ROWS * NCLASS) >> 16;                       // dim1_stride[47:16]

    asm volatile("tensor_load_to_lds %0, %1" :: "s"(g0), "s"(g1) : "memory");
}

// wave32 max + first-occurrence argmax over one row (lane holds elems 4L..4L+3)
__device__ __forceinline__ void row_argmax(f4 v, int lane, float& m_out, int& idx_out) {
    float m = fmaxf(fmaxf(v.x, v.y), fmaxf(v.z, v.w));
#pragma unroll
    for (int off = 16; off >= 1; off >>= 1)
        m = fmaxf(m, __shfl_xor(m, off, 32));
    int cand = 0x7FFFFFFF;
    if      (v.x == m) cand = lane * 4;
    else if (v.y == m) cand = lane * 4 + 1;
    else if (v.z == m) cand = lane * 4 + 2;
    else if (v.w == m) cand = lane * 4 + 3;
    unsigned long long bm = __ballot(cand != 0x7FFFFFFF);
    int fl = __ffsll(bm) - 1;                 // lowest lane holding the max
    idx_out = __shfl(cand, fl, 32);
    m_out   = m;
}

__device__ __forceinline__ void bin_accum(float* bins, float m, int idx, int label) {
    int b = (int)ceilf(m * (float)N_BINS) - 1;
    b = b < 0 ? 0 : (b > N_BINS - 1 ? N_BINS - 1 : b);
    atomicAdd(&bins[b], 1.0f);
    atomicAdd(&bins[N_BINS + b], m);
    if (idx == label) atomicAdd(&bins[2 * N_BINS + b], 1.0f);
}

__global__ void rd_init_ws(float* __restrict__ ws) {
    if (threadIdx.x < 3 * N_BINS) ws[threadIdx.x] = 0.0f;
}

__global__ __launch_bounds__(BLOCK_THREADS) void rd_main(const float* __restrict__ sm,
                                                         const int* __restrict__ labels,
                                                         float* __restrict__ ws,
                                                         int ntiles) {
    __shared__ __align__(16) float tbuf[2 * TILE_ROWS * NCLASS];   // 48KB double buffer
    __shared__ float s_bins[3 * N_BINS];
    if (threadIdx.x < 3 * N_BINS) s_bins[threadIdx.x] = 0.0f;

    const int      lane     = threadIdx.x & 31;
    const int      wid      = __builtin_amdgcn_readfirstlane(threadIdx.x) >> 5; // scalar wave id
    const uint32_t lds_base = (uint32_t)(uintptr_t)&tbuf[0];   // flat LDS addr low 32 = LDS offset
    const int      t0       = blockIdx.x;
    const int      step     = gridDim.x;

    if (wid == 0 && t0 < ntiles) tdm_issue(sm, t0, lds_base, 0);   // prologue: tile t0 -> buf0
    __syncthreads();                                               // also publishes s_bins init

    int it = 0;
    for (int t = t0; t < ntiles; t += step, ++it) {
        if (wid == 0) {
            if (t + step < ntiles) {
                tdm_issue(sm, t + step, lds_base, (it + 1) & 1);   // overlap next tile
                __builtin_amdgcn_s_wait_tensorcnt(1);              // in-order => tile t done
            } else {
                __builtin_amdgcn_s_wait_tensorcnt(0);
            }
        }
        __syncthreads();                                           // buf[it&1] ready for all waves

        const float* buf = &tbuf[(size_t)(it & 1) * TILE_ROWS * NCLASS];
#pragma unroll
        for (int rr = 0; rr < ROWS_PER_WAVE; ++rr) {
            const int lrow = wid * ROWS_PER_WAVE + rr;
            f4 v = *((const f4*)(buf + (size_t)lrow * NCLASS) + lane);   // ds_load_b128
            float m; int idx;
            row_argmax(v, lane, m, idx);
            if (lane == 0) {
                long long grow = (long long)t * TILE_ROWS + lrow;
                bin_accum(s_bins, m, idx, labels[grow]);           // ds_add_f32
            }
        }
        __syncthreads();            // all reads of buf[it&1] done before it is re-filled
    }

    __syncthreads();
    if (threadIdx.x < 3 * N_BINS) {
        float val = s_bins[threadIdx.x];
        if (val != 0.0f) atomicAdd(&ws[threadIdx.x], val);         // global_atomic_add_f32
    }
}

// Tail: rows [row0, N) (< TILE_ROWS of them) with direct NT loads.
__global__ __launch_bounds__(BLOCK_THREADS) void rd_tail(const float* __restrict__ sm,
                                                         const int* __restrict__ labels,
                                                         float* __restrict__ ws,
                                                         int row0, int N) {
    const int lane = threadIdx.x & 31;
    const int wave = threadIdx.x >> 5;
    for (int row = row0 + wave; row < N; row += WAVES_PER_BLOCK) {
        f4 v = __builtin_nontemporal_load((const f4*)(sm + (size_t)row * NCLASS) + lane);
        float m; int idx;
        row_argmax(v, lane, m, idx);
        if (lane == 0) bin_accum(ws, m, idx, labels[row]);         // few rows: global atomics ok
    }
}

__global__ void rd_final(const float* __restrict__ ws, float* __restrict__ out) {
    int b = threadIdx.x;
    if (b < N_BINS) {
        float c    = ws[b];
        float conf = ws[N_BINS + b];
        float acc  = ws[2 * N_BINS + b];
        out[b]          = (c > 0.0f) ? conf / c : 0.0f;   // x: avg confidence per bin
        out[N_BINS + b] = (c > 0.0f) ? acc / c  : 0.0f;   // y: accuracy per bin
    }
}

extern "C" void kernel_launch(void* const* d_in, const int* in_sizes, int n_in,
                              void* d_out, int out_size, void* d_ws, size_t ws_size,
                              hipStream_t stream) {
    const float* sm     = (const float*)d_in[0];   // [N, 128] float32
    const int*   labels = (const int*)d_in[1];     // [N] int
    const int    N      = in_sizes[1];
    float*       ws     = (float*)d_ws;            // 30 floats: cnt/conf/acc per bin
    float*       out    = (float*)d_out;           // 20 floats: x[10] ++ y[10]

    const int ntiles = N / TILE_ROWS;
    const int row0   = ntiles * TILE_ROWS;

    rd_init_ws<<<1, 64, 0, stream>>>(ws);
    if (ntiles > 0) {
        int grid = ntiles < 2048 ? ntiles : 2048;
        rd_main<<<grid, BLOCK_THREADS, 0, stream>>>(sm, labels, ws, ntiles);
    }
    if (row0 < N) rd_tail<<<1, BLOCK_THREADS, 0, stream>>>(sm, labels, ws, row0, N);
    rd_final<<<1, 32, 0, stream>>>(ws, out);
}